// YOLO_loss_44152263803093
// MI455X (gfx1250) — compile-verified
//
#include <hip/hip_runtime.h>

// ---------------- types / helpers ----------------
typedef __attribute__((ext_vector_type(2))) float v2f;
typedef __attribute__((ext_vector_type(8))) float v8f;
typedef int v2i __attribute__((vector_size(2 * sizeof(int))));

#define GLOBAL_AS __attribute__((address_space(1)))
#define LDS_AS    __attribute__((address_space(3)))

__device__ __forceinline__ float sq(float x) { return x * x; }

// Async global->LDS copy of one float2 element (8 bytes), per lane.
// Tracked by ASYNCcnt; complete with async_wait_all() before reading LDS.
// Builtin signature (from clang diagnostic): (v2i AS1*, v2i AS3*, Ii offset, Ii cpol)
__device__ __forceinline__ void async_copy_f2(const float2* gsrc, float2* ldst) {
#if __has_builtin(__builtin_amdgcn_global_load_async_to_lds_b64)
  __builtin_amdgcn_global_load_async_to_lds_b64(
      (GLOBAL_AS v2i*)gsrc, (LDS_AS v2i*)ldst, /*offset=*/0, /*cpol=*/0);
#else
  unsigned lds_off = (unsigned)(__UINTPTR_TYPE__)(LDS_AS void*)ldst;
  unsigned long long ga = (unsigned long long)(__UINTPTR_TYPE__)gsrc;
  asm volatile("global_load_async_to_lds_b64 %0, %1, off"
               :
               : "v"(lds_off), "v"(ga)
               : "memory");
#endif
}

__device__ __forceinline__ void async_wait_all() {
#if __has_builtin(__builtin_amdgcn_s_wait_asynccnt)
  __builtin_amdgcn_s_wait_asynccnt(0);
#else
  asm volatile("s_wait_asynccnt 0" ::: "memory");
#endif
}

// Exact fp32 wave(32)-wide sum using V_WMMA_F32_16X16X4_F32.
// Stage 1: D = A(p in K=0/K=2 cols) x Ones(4x16)  -> D[m][n] = p[m] + p[m+16]
// Stage 2: 4 accumulating WMMAs with A = Ones sum the 16 row-sums column-wise.
// Requires EXEC all ones: call only from uniform control flow with full block.
__device__ __forceinline__ float wave_reduce_sum(float p) {
#if __has_builtin(__builtin_amdgcn_wmma_f32_16x16x4_f32)
  v2f a;   a[0] = p;      a[1] = 0.0f;
  v2f one; one[0] = 1.0f; one[1] = 1.0f;
  v8f z = {};
  v8f d = __builtin_amdgcn_wmma_f32_16x16x4_f32(false, a, false, one, (short)0, z,
                                                false, false);
  v8f acc = {};
#pragma unroll
  for (int v = 0; v < 8; v += 2) {
    v2f bb; bb[0] = d[v]; bb[1] = d[v + 1];
    acc = __builtin_amdgcn_wmma_f32_16x16x4_f32(false, one, false, bb, (short)0, acc,
                                                false, false);
  }
  return acc[0];  // full wave sum, broadcast to every lane
#else
#pragma unroll
  for (int m = 16; m >= 1; m >>= 1) p += __shfl_xor(p, m, 32);
  return p;
#endif
}

// ---------------- kernel 1: per-batch-item partial loss ----------------
// One 64-thread block (2 waves) per batch item. Stages the contiguous
// [30][7][7] tiles of pred and label (5880 B each, 8-byte aligned) into LDS
// via async global->LDS b64 transfers, then 49 lanes evaluate one cell each.
__global__ __launch_bounds__(64) void yolo_partial_kernel(
    const float* __restrict__ pred, const float* __restrict__ label,
    float* __restrict__ partials) {
  __shared__ __align__(16) float sp[30 * 49];
  __shared__ __align__(16) float sl[30 * 49];
  __shared__ float swave[2];

  const int b = blockIdx.x;
  const int t = threadIdx.x;

  // 1470 floats = 735 float2 per tensor; base offset 5880*b is 8B-aligned.
  const float2* gp2 = (const float2*)(pred + (size_t)b * 1470);
  const float2* gl2 = (const float2*)(label + (size_t)b * 1470);
  float2* sp2 = (float2*)sp;
  float2* sl2 = (float2*)sl;
#pragma unroll 4
  for (int k = t; k < 735; k += 64) {
    async_copy_f2(gp2 + k, sp2 + k);
    async_copy_f2(gl2 + k, sl2 + k);
  }
  async_wait_all();   // this wave's async transfers landed in LDS
  __syncthreads();    // both waves' transfers visible

  const bool active = (t < 49);
  const int tc = active ? t : 0;          // keep all 64 lanes converged
  const float fi = (float)(tc / 7);       // paired with x (axis 2)
  const float fj = (float)(tc % 7);       // paired with y (axis 3)
  const float Gf = (float)(1.0 / 7.0);

  float P[30], L[30];
#pragma unroll
  for (int c = 0; c < 30; ++c) {
    P[c] = sp[c * 49 + tc];
    L[c] = sl[c * 49 + tc];
  }

  // degenerate label "box" (point)
  const float lx = (L[0] + fi) * Gf - 0.5f * L[2];
  const float ly = (L[1] + fj) * Gf - 0.5f * L[3];

  auto iou_vs_point = [&](float x, float y, float w, float h) {
    float cx = (x + fi) * Gf, cy = (y + fj) * Gf;
    float x1 = cx - 0.5f * w, x2 = cx + 0.5f * w;
    float y1 = cy - 0.5f * h, y2 = cy + 0.5f * h;
    float ix1 = fmaxf(x1, lx), ix2 = fminf(x2, lx);
    float iy1 = fmaxf(y1, ly), iy2 = fminf(y2, ly);
    float inter = fmaxf(ix2 - ix1, 0.0f) * fmaxf(iy2 - iy1, 0.0f);
    float area_a = (x2 - x1) * (y2 - y1);  // area_b == 0
    float uni = area_a - inter;
    return inter / (uni == 0.0f ? 1.0f : uni);
  };

  const float iou1 = iou_vs_point(P[0], P[1], P[2], P[3]);
  const float iou2 = iou_vs_point(P[5], P[6], P[7], P[8]);
  const bool sel1 = iou1 > iou2;
  const bool obj = (L[4] == 1.0f);

  const float t1 = sq(P[0] - L[0]) + sq(P[1] - L[1]) +
                   sq(sqrtf(P[2]) - sqrtf(L[2])) + sq(sqrtf(P[3]) - sqrtf(L[3]));
  const float t2 = sq(P[5] - L[5]) + sq(P[6] - L[6]) +
                   sq(sqrtf(P[7]) - sqrtf(L[7])) + sq(sqrtf(P[8]) - sqrtf(L[8]));

  const float obj_term    = sel1 ? t1 : t2;
  const float conf_term   = sel1 ? sq(P[4] - 1.0f) : sq(P[9] - 1.0f);
  const float noobj_obj   = sel1 ? sq(P[9]) : sq(P[4]);
  const float noobj_empty = sq(P[4]) + sq(P[9]);

  float cls = 0.0f;
#pragma unroll
  for (int c = 10; c < 30; ++c) cls += sq(P[c] - L[c]);

  // obj_weight = noobj_weight = 0.5
  const float cell = obj ? (0.5f * (obj_term + noobj_obj) + conf_term + cls)
                         : (0.5f * noobj_empty);
  const float partial = active ? cell : 0.0f;

  // wave-level exact fp32 reduction via WMMA (uniform flow, EXEC all ones)
  const float wsum = wave_reduce_sum(partial);
  if ((t & 31) == 0) swave[t >> 5] = wsum;
  __syncthreads();
  if (t == 0) partials[b] = swave[0] + swave[1];
}

// ---------------- kernel 2: deterministic final reduction ----------------
__global__ __launch_bounds__(256) void yolo_reduce_kernel(
    const float* __restrict__ partials, float* __restrict__ out, int n) {
  __shared__ float s[8];
  const int t = threadIdx.x;
  float acc = 0.0f;
  for (int k = t; k < n; k += 256) acc += partials[k];  // fixed order
  const float wsum = wave_reduce_sum(acc);
  if ((t & 31) == 0) s[t >> 5] = wsum;
  __syncthreads();
  if (t == 0) {
    float tot = 0.0f;
#pragma unroll
    for (int w = 0; w < 8; ++w) tot += s[w];
    out[0] = tot / (float)n;  // reference divides by B
  }
}

// ---------------- launch ----------------
extern "C" void kernel_launch(void* const* d_in, const int* in_sizes, int n_in,
                              void* d_out, int out_size, void* d_ws, size_t ws_size,
                              hipStream_t stream) {
  const float* pred  = (const float*)d_in[0];
  const float* label = (const float*)d_in[1];
  float* out = (float*)d_out;
  float* partials = (float*)d_ws;  // B floats of scratch

  const int B = in_sizes[0] / (30 * 7 * 7);

  yolo_partial_kernel<<<B, 64, 0, stream>>>(pred, label, partials);
  yolo_reduce_kernel<<<1, 256, 0, stream>>>(partials, out, B);
}